// TicRows_22342419874079
// MI455X (gfx1250) — compile-verified
//
#include <hip/hip_runtime.h>

// ---------------------------------------------------------------------------
// TicRows: out = x @ W + b, where W's column-block r (256 cols) only has
// 192 nonzero rows (3 board cells x 64 channels). We run 49x2 structured
// GEMM blocks with native fp32 WMMA (v_wmma_f32_16x16x4_f32) on gfx1250.
// Software-pipelined B fragments (LDS) + vectorized b128 store epilogue
// through an in-LDS transpose (output write is the roofline term: ~411 MB).
// A fragments are preloaded in chunks of 4 to keep VGPR usage < 256
// (avoids s_set_vgpr_msb toggling and keeps occupancy at the LDS cap).
// ---------------------------------------------------------------------------

#define TR_BATCH 8192
#define TR_KTOT  1728      // 27 * 64, x's inner dim
#define TR_ROWS  49
#define TR_CH    256
#define TR_NOUT  (TR_ROWS * TR_CH)   // 12544
#define TR_MBLK  128
#define TR_NBLK  128
#define TR_KEFF  192       // 3 cells * 64 channels
#define TR_TS    (TR_NBLK + 4)       // padded LDS row stride for transpose (132)

typedef __attribute__((ext_vector_type(2))) float v2f;
typedef __attribute__((ext_vector_type(4))) float v4f;
typedef __attribute__((ext_vector_type(8))) float v8f;

// ---- constexpr port of init_wincon_matrix(): cells[r][e] = board index ----
struct CellTab { int c[TR_ROWS][3]; int ncol; };

constexpr CellTab make_cells() {
    CellTab t{};
    int col = 0, step = 0;
    int x = 0, y = 0, z = 0;
#define TIC_PUT() { t.c[col][step] = x + 3 * y + 9 * z; step = (step + 1) % 3; }
    for (x = 0; x < 3; ++x) {
        for (y = 0; y < 3; ++y) { for (z = 0; z < 3; ++z) TIC_PUT(); ++col; }
        for (z = 0; z < 3; ++z) { for (y = 0; y < 3; ++y) TIC_PUT(); ++col; }
        for (y = 0; y < 3; ++y) { z = y;     TIC_PUT(); } ++col;
        for (y = 0; y < 3; ++y) { z = 2 - y; TIC_PUT(); } ++col;
    }
    for (z = 0; z < 3; ++z) {
        for (y = 0; y < 3; ++y) { for (x = 0; x < 3; ++x) TIC_PUT(); ++col; }
        for (y = 0; y < 3; ++y) { x = y;     TIC_PUT(); } ++col;
        for (y = 0; y < 3; ++y) { x = 2 - y; TIC_PUT(); } ++col;
    }
    for (y = 0; y < 3; ++y) {
        for (z = 0; z < 3; ++z) { x = z;     TIC_PUT(); } ++col;
        for (z = 0; z < 3; ++z) { x = 2 - z; TIC_PUT(); } ++col;
    }
    for (x = 0; x < 3; ++x) { y = x;     z = x;     TIC_PUT(); } ++col;
    for (x = 0; x < 3; ++x) { y = 2 - x; z = 2 - x; TIC_PUT(); } ++col;
    for (x = 0; x < 3; ++x) { y = x;     z = 2 - x; TIC_PUT(); } ++col;
    for (x = 0; x < 3; ++x) { z = x;     y = 2 - x; TIC_PUT(); } ++col;
#undef TIC_PUT
    t.ncol = col;
    return t;
}

static_assert(make_cells().ncol == TR_ROWS, "win-line generator produced wrong column count");
static_assert(make_cells().c[0][0] == 0 && make_cells().c[0][1] == 9 && make_cells().c[0][2] == 18,
              "win-line generator mismatch on first line (x=0,y=0,z=0..2)");

__constant__ CellTab TR_CELLS = make_cells();

// ---------------------------------------------------------------------------
// One workgroup (8 waves, 256 threads) computes a 128x128 output tile.
//   r      = blockIdx.x >> 1  (row-group, selects 3 cells + v slice)
//   nhalf  = blockIdx.x & 1   (which 128-col half of the 256-col group)
//   m0     = blockIdx.y * 128
// Phase 1: stage B = w_sub (192x128 slice of v) in LDS, K-pair swizzled so a
//          lane's WMMA B fragment {K=k, K=k+1 @ same N} is one ds_load_b64.
// Phase 2: K loop, software-pipelined (B for step k+1 loads during WMMAs of k;
//          A preloaded 4 fragments at a time).
// Phase 3: barrier, reuse LDS to transpose accumulators -> dense b128 stores.
// ---------------------------------------------------------------------------
__global__ __launch_bounds__(256)
void tic_rows_wmma_f32(const float* __restrict__ x,
                       const float* __restrict__ v,
                       const float* __restrict__ bias,
                       float* __restrict__ out) {
    extern __shared__ float lds[];   // 24576 floats = 96 KB (B stage, then transpose)

    const int r     = blockIdx.x >> 1;
    const int nhalf = blockIdx.x & 1;
    const int col0  = r * TR_CH + nhalf * TR_NBLK;   // global output column base
    const int m0    = blockIdx.y * TR_MBLK;

    const int tid   = threadIdx.x;
    const int lane  = tid & 31;
    const int wave  = tid >> 5;
    const int lhalf = lane >> 4;    // 0: lanes 0-15, 1: lanes 16-31
    const int l16   = lane & 15;

    // ---- Phase 1: stage B. lds[(k>>1)*256 + 2*n + (k&1)] = v[k_row, r, half+n]
    for (int k = wave; k < TR_KEFF; k += 8) {
        const v4f* src = (const v4f*)(v + ((size_t)k * TR_ROWS + r) * TR_CH + nhalf * TR_NBLK) + lane;
        v4f d = *src;
        const int n = lane * 4;
        float* dst = lds + (k >> 1) * (2 * TR_NBLK) + (k & 1);
        dst[2 * n + 0] = d.x;
        dst[2 * n + 2] = d.y;
        dst[2 * n + 4] = d.z;
        dst[2 * n + 6] = d.w;
    }
    __syncthreads();

    // ---- Phase 2: K loop.
    // A layout (16x4 f32): lanes 0-15 K={k,k+1}, lanes 16-31 K={k+2,k+3}, M=lane%16.
    const int arow = m0 + wave * 16 + l16;
    const float* xrow = x + (size_t)arow * TR_KTOT;
    const float* ldsb = lds + (size_t)l16 * 2;   // per-lane N offset into B rows

    v8f acc[8];
    const v8f vzero = {0.f, 0.f, 0.f, 0.f, 0.f, 0.f, 0.f, 0.f};
#pragma unroll
    for (int t = 0; t < 8; ++t) acc[t] = vzero;

#pragma unroll
    for (int e = 0; e < 3; ++e) {
        const float* xseg = xrow + TR_CELLS.c[r][e] * 64;   // contiguous 64-col slice

        // Preload B fragments for k-step 0 of this cell.
        v2f bf[8], bfn[8];
#pragma unroll
        for (int t = 0; t < 8; ++t)
            bf[t] = *(const v2f*)(ldsb + (size_t)(e * 32 + lhalf) * (2 * TR_NBLK) + t * 32);

#pragma unroll
        for (int kg = 0; kg < 4; ++kg) {
            // Preload 4 A fragments (one clause of b64 loads, only 8 VGPRs live).
            v2f av[4];
#pragma unroll
            for (int i = 0; i < 4; ++i)
                av[i] = *(const v2f*)(xseg + (kg * 4 + i) * 4 + lhalf * 2);

#pragma unroll
            for (int i = 0; i < 4; ++i) {
                const int k4 = kg * 4 + i;
                // Software pipeline: fetch step k4+1's B before step k4's WMMAs.
                if (k4 + 1 < 16) {
#pragma unroll
                    for (int t = 0; t < 8; ++t)
                        bfn[t] = *(const v2f*)(ldsb +
                            (size_t)(e * 32 + (k4 + 1) * 2 + lhalf) * (2 * TR_NBLK) + t * 32);
                }
#pragma unroll
                for (int t = 0; t < 8; ++t)
                    acc[t] = __builtin_amdgcn_wmma_f32_16x16x4_f32(
                        /*neg_a=*/false, av[i], /*neg_b=*/false, bf[t],
                        /*c_mod=*/(short)0, acc[t],
                        /*reuse_a=*/false, /*reuse_b=*/false);
                if (k4 + 1 < 16) {
#pragma unroll
                    for (int t = 0; t < 8; ++t) bf[t] = bfn[t];   // register rename
                }
            }
        }
    }

    // ---- Phase 3: epilogue. Reuse LDS (all waves finished reading B) to
    // transpose each wave's 16x128 tile, then emit dense 512B/instruction
    // global_store_b128 with a vectorized bias add.
    __syncthreads();

    // C/D 16x16 f32 layout: VGPR j -> rows {j (lanes 0-15), j+8 (lanes 16-31)},
    // col = lane%16. Scatter into per-wave-private padded LDS region.
    const int rbase = wave * 16 + lhalf * 8;
#pragma unroll
    for (int t = 0; t < 8; ++t) {
#pragma unroll
        for (int j = 0; j < 8; ++j)
            lds[(rbase + j) * TR_TS + t * 16 + l16] = acc[t][j];
    }
    // Per-wave DS ops are in-order: our own stores are visible to our loads.
    const v4f bvec = *(const v4f*)(bias + col0 + lane * 4);
#pragma unroll
    for (int j2 = 0; j2 < 16; ++j2) {
        const int rl = wave * 16 + j2;
        v4f val = *(const v4f*)(lds + rl * TR_TS + lane * 4);
        val = val + bvec;
        *(v4f*)(out + (size_t)(m0 + rl) * TR_NOUT + col0 + lane * 4) = val;
    }
}

extern "C" void kernel_launch(void* const* d_in, const int* in_sizes, int n_in,
                              void* d_out, int out_size, void* d_ws, size_t ws_size,
                              hipStream_t stream) {
    (void)in_sizes; (void)n_in; (void)out_size; (void)d_ws; (void)ws_size;
    const float* x    = (const float*)d_in[0];
    const float* v    = (const float*)d_in[1];
    const float* bias = (const float*)d_in[2];
    float* out        = (float*)d_out;

    dim3 grid(TR_ROWS * 2, TR_BATCH / TR_MBLK);   // 98 x 64
    dim3 block(256);
    const size_t lds_bytes = (TR_KEFF / 2) * (2 * TR_NBLK) * sizeof(float);  // 96 KB
    tic_rows_wmma_f32<<<grid, block, lds_bytes, stream>>>(x, v, bias, out);
}